// PatternLSTM_40329742909504
// MI455X (gfx1250) — compile-verified
//
#include <hip/hip_runtime.h>
#include <math.h>

typedef float v2f __attribute__((ext_vector_type(2)));
typedef float v4f __attribute__((ext_vector_type(4)));
typedef float v8f __attribute__((ext_vector_type(8)));

#define GDIM 4096   // 4*HID gate dimension
#define HIDN 1024
#define SEQN 16

// out[16][4096] = A[16][K] * W[4096][K]^T + b1 + b2
// One block per 16-wide N tile; K split across the 8 waves; LDS reduction.
// K must be a multiple of 128 (2048 and 1024 both are).
__global__ __launch_bounds__(256) void gemm16_wmma(
    const float* __restrict__ A, const float* __restrict__ W,
    const float* __restrict__ b1, const float* __restrict__ b2,
    float* __restrict__ out, int K)
{
    __shared__ float red[8][32][8];   // 8 KB
    const int tid  = threadIdx.x;
    const int wave = tid >> 5;
    const int lane = tid & 31;
    const int m    = lane & 15;       // A row (M) for A-frag, N column for B-frag
    const int hh   = lane >> 4;       // lane-half -> K sub-offset select
    const int n0   = blockIdx.x << 4; // output column tile base
    const int kslice = K >> 3;        // per-wave K chunk (256 or 128; multiple of 16)
    const int kbeg = wave * kslice;
    const int kend = kbeg + kslice;

    const float* Arow = A + m * K + 4 * hh;        // A[m][* + 4h]
    const float* Wrow = W + (n0 + m) * K + 4 * hh; // W[n0+n][* + 4h], n = lane&15

    v8f acc = {};
    for (int k = kbeg; k < kend; k += 16) {
        // b128 loads: lane covers K {k+4h..k+4h+3} and {k+8+4h..k+8+4h+3}
        v4f a0 = *(const v4f*)(Arow + k);
        v4f w0 = *(const v4f*)(Wrow + k);
        v4f a1 = *(const v4f*)(Arow + k + 8);
        v4f w1 = *(const v4f*)(Wrow + k + 8);
        // Each WMMA reduces a consistent 4-element K subset (A/B frags agree).
        v2f av, wv;
        av.x = a0.x; av.y = a0.y; wv.x = w0.x; wv.y = w0.y;
        acc = __builtin_amdgcn_wmma_f32_16x16x4_f32(false, av, false, wv, (short)0, acc, false, false);
        av.x = a0.z; av.y = a0.w; wv.x = w0.z; wv.y = w0.w;
        acc = __builtin_amdgcn_wmma_f32_16x16x4_f32(false, av, false, wv, (short)0, acc, false, false);
        av.x = a1.x; av.y = a1.y; wv.x = w1.x; wv.y = w1.y;
        acc = __builtin_amdgcn_wmma_f32_16x16x4_f32(false, av, false, wv, (short)0, acc, false, false);
        av.x = a1.z; av.y = a1.w; wv.x = w1.z; wv.y = w1.w;
        acc = __builtin_amdgcn_wmma_f32_16x16x4_f32(false, av, false, wv, (short)0, acc, false, false);
    }

#pragma unroll
    for (int e = 0; e < 8; ++e) red[wave][lane][e] = acc[e];
    __syncthreads();

    // 256 threads: one per (lane, elem); sum the 8 partial tiles, add bias, store.
    const int lane_t = tid >> 3;            // 0..31
    const int e      = tid & 7;             // 0..7
    float s = 0.f;
#pragma unroll
    for (int w = 0; w < 8; ++w) s += red[w][lane_t][e];
    const int n  = lane_t & 15;
    const int mr = e + ((lane_t >> 4) << 3);    // C row (sequence index)
    const int g  = n0 + n;                      // gate index
    out[mr * GDIM + g] = s + b1[g] + b2[g];
}

// One LSTM time step. 64 blocks x 256 threads; block b owns hidden units
// j in [16b, 16b+16) -> 64 rows of Whh (i,f,g,o), each row read once, coalesced.
__global__ __launch_bounds__(256) void lstm_step(
    const float* __restrict__ xg,     // [16][4096] precomputed input gates (+bias)
    const float* __restrict__ Whh,    // [4096][1024]
    const float* __restrict__ hprev,  // [1024] previous hidden state (read-only)
    float* __restrict__ cbuf,         // [1024] cell state (in-place, per-j private)
    float* __restrict__ hout,         // [1024] h_t destination
    float* __restrict__ last,         // nullable: also write h_t here (last step)
    int t)
{
    __shared__ float h_s[HIDN];
    __shared__ float gates_s[64];
    const int tid  = threadIdx.x;
    const int wave = tid >> 5;
    const int lane = tid & 31;
    const int jbase = blockIdx.x << 4;

    // stage h in LDS (b128 copies)
    for (int i = tid; i < HIDN / 4; i += 256)
        *(v4f*)&h_s[i * 4] = *(const v4f*)&hprev[i * 4];
    __syncthreads();

    // 64 dot products of length 1024: wave w handles d = w, w+8, ..., w+56
    for (int d = wave; d < 64; d += 8) {
        const int jl   = d & 15;
        const int gate = d >> 4;
        const int row  = jbase + jl + gate * HIDN;
        const v4f* wr4 = (const v4f*)(Whh + row * HIDN);
        float s = 0.f;
#pragma unroll
        for (int k4 = 0; k4 < HIDN / 4; k4 += 32) {
            v4f w4 = wr4[k4 + lane];
            v4f h4 = *(const v4f*)&h_s[(k4 + lane) * 4];
            s += w4.x * h4.x + w4.y * h4.y + w4.z * h4.z + w4.w * h4.w;
        }
#pragma unroll
        for (int off = 16; off > 0; off >>= 1) s += __shfl_down(s, off, 32);
        if (lane == 0) gates_s[d] = s + xg[t * GDIM + row];
    }
    __syncthreads();

    if (tid < 16) {
        const int j = jbase + tid;
        float i_ = gates_s[tid];
        float f_ = gates_s[16 + tid];
        float g_ = gates_s[32 + tid];
        float o_ = gates_s[48 + tid];
        i_ = 1.f / (1.f + expf(-i_));
        f_ = 1.f / (1.f + expf(-f_));
        g_ = tanhf(g_);
        o_ = 1.f / (1.f + expf(-o_));
        const float c = f_ * cbuf[j] + i_ * g_;
        cbuf[j] = c;
        const float h = o_ * tanhf(c);
        hout[j] = h;
        if (last) last[j] = h;
    }
}

__global__ void zero_buf(float* __restrict__ p, int n)
{
    const int i = blockIdx.x * blockDim.x + threadIdx.x;
    if (i < n) p[i] = 0.f;
}

extern "C" void kernel_launch(void* const* d_in, const int* in_sizes, int n_in,
                              void* d_out, int out_size, void* d_ws, size_t ws_size,
                              hipStream_t stream) {
    const float* x     = (const float*)d_in[0];   // [1,16,2048]
    const float* w_ih0 = (const float*)d_in[1];   // [4096,2048]
    const float* w_hh0 = (const float*)d_in[2];   // [4096,1024]
    const float* b_ih0 = (const float*)d_in[3];   // [4096]
    const float* b_hh0 = (const float*)d_in[4];   // [4096]
    const float* w_ih1 = (const float*)d_in[5];   // [4096,1024]
    const float* w_hh1 = (const float*)d_in[6];   // [4096,1024]
    const float* b_ih1 = (const float*)d_in[7];   // [4096]
    const float* b_hh1 = (const float*)d_in[8];   // [4096]
    float* out = (float*)d_out;                   // [1024 last] ++ [16*1024 h2]

    // Workspace layout (floats): xg[65536] | h1[16384] | hz[1024] | c[1024]
    float* ws = (float*)d_ws;
    float* xg = ws;
    float* h1 = ws + SEQN * GDIM;                 // 65536
    float* hz = h1 + SEQN * HIDN;                 // +16384
    float* cb = hz + HIDN;                        // +1024
    float* h2 = out + HIDN;                       // h2 region of d_out

    // ---- Layer 0 ----
    zero_buf<<<8, 256, 0, stream>>>(hz, 2 * HIDN);            // zero h0, c0
    gemm16_wmma<<<GDIM / 16, 256, 0, stream>>>(x, w_ih0, b_ih0, b_hh0, xg, 2048);
    for (int t = 0; t < SEQN; ++t) {
        const float* hprev = (t == 0) ? hz : (h1 + (t - 1) * HIDN);
        lstm_step<<<HIDN / 16, 256, 0, stream>>>(xg, w_hh0, hprev, cb,
                                                 h1 + t * HIDN, nullptr, t);
    }

    // ---- Layer 1 ----
    zero_buf<<<8, 256, 0, stream>>>(hz, 2 * HIDN);            // re-zero h0, c0
    gemm16_wmma<<<GDIM / 16, 256, 0, stream>>>(h1, w_ih1, b_ih1, b_hh1, xg, 1024);
    for (int t = 0; t < SEQN; ++t) {
        const float* hprev = (t == 0) ? hz : (h2 + (t - 1) * HIDN);
        lstm_step<<<HIDN / 16, 256, 0, stream>>>(xg, w_hh1, hprev, cb,
                                                 h2 + t * HIDN,
                                                 (t == SEQN - 1) ? out : nullptr, t);
    }
}